// GNN_ATT_DDI_3367254360366
// MI455X (gfx1250) — compile-verified
//
#include <hip/hip_runtime.h>
#include <math.h>

#define BS 16
#define C 256
#define T 768
#define PATCH 12
#define NPATCH 64
#define FF 256
#define HEADS 4
#define HD 64
#define ALPHA 0.5f
// 1/sqrt(1+1e-5): eval-mode BN scale with running stats mean=0, var=1
#define RSQ 0.99999500003749966f

typedef __attribute__((ext_vector_type(2))) float v2f;
typedef __attribute__((ext_vector_type(8))) float v8f;

__device__ __forceinline__ float gelu_exact(float x) {
  return 0.5f * x * (1.0f + erff(x * 0.70710678118654752f));
}

// ---------------------------------------------------------------------------
// One-time folding of the rank-1 embedding through the projection weights.
//   uq = Wq@We, cq = Wq@be+bq (same k,v);  P_h = <uq,uk>_h/8, R_h = <cq,uk>_h/8
//   M1[h][f] = sum_{d in head h} Wm1[f,d]*uv[d];  base = Wm1@cv + bm1
// ---------------------------------------------------------------------------
__global__ void k_precompute(const float* __restrict__ We, const float* __restrict__ be,
                             const float* __restrict__ Wq, const float* __restrict__ bq,
                             const float* __restrict__ Wk, const float* __restrict__ bk,
                             const float* __restrict__ Wv, const float* __restrict__ bv,
                             const float* __restrict__ Wm1, const float* __restrict__ bm1,
                             float* __restrict__ M1g, float* __restrict__ baseg,
                             float* __restrict__ PRg) {
  __shared__ float uq[FF], uk[FF], uv[FF], cq[FF], ck[FF], cv[FF];
  int f = threadIdx.x;
  const float* wqr = Wq + f * FF;
  const float* wkr = Wk + f * FF;
  const float* wvr = Wv + f * FF;
  float squ = 0.f, scq = 0.f, sku = 0.f, skc = 0.f, svu = 0.f, svc = 0.f;
  for (int j = 0; j < FF; ++j) {
    float e = We[j], bb = be[j];
    squ += wqr[j] * e; scq += wqr[j] * bb;
    sku += wkr[j] * e; skc += wkr[j] * bb;
    svu += wvr[j] * e; svc += wvr[j] * bb;
  }
  uq[f] = squ; cq[f] = scq + bq[f];
  uk[f] = sku; ck[f] = skc + bk[f];
  uv[f] = svu; cv[f] = svc + bv[f];
  (void)ck;
  __syncthreads();
  const float* wmr = Wm1 + f * FF;
  for (int h = 0; h < HEADS; ++h) {
    float s = 0.f;
    for (int d = 0; d < HD; ++d) s += wmr[h * HD + d] * uv[h * HD + d];
    M1g[h * FF + f] = s;
  }
  float sb = bm1[f];
  for (int j = 0; j < FF; ++j) sb += wmr[j] * cv[j];
  baseg[f] = sb;
  if (f < HEADS) {
    float sp = 0.f, sr = 0.f;
    for (int d = 0; d < HD; ++d) {
      sp += uq[f * HD + d] * uk[f * HD + d];
      sr += cq[f * HD + d] * uk[f * HD + d];
    }
    PRg[f] = sp * 0.125f;          // 1/sqrt(HD=64)
    PRg[HEADS + f] = sr * 0.125f;
  }
}

// out patch 0 = BN0(x) patch 0
__global__ void k_init(const float* __restrict__ x, const float* __restrict__ g0,
                       const float* __restrict__ b0, float* __restrict__ out) {
  int idx = blockIdx.x * blockDim.x + threadIdx.x;  // (b*C+c)*PATCH + p
  int p = idx % PATCH;
  int bc = idx / PATCH;
  int c = bc % C;
  int gi = c * T + p;
  size_t xi = (size_t)bc * T + p;
  out[xi] = x[xi] * (g0[gi] * RSQ) + b0[gi];
}

// BN1(prev) @ W_agg.T -> gelu -> +BN0(x_t) = res ; t = BN2(res) transposed [b,p,c]
__global__ void k_stepA(const float* __restrict__ x,
                        const float* __restrict__ g0, const float* __restrict__ b0,
                        const float* __restrict__ g1, const float* __restrict__ b1,
                        const float* __restrict__ g2, const float* __restrict__ b2,
                        const float* __restrict__ Wagg, const float* __restrict__ bagg,
                        const float* __restrict__ out, float* __restrict__ res,
                        float* __restrict__ t, int step) {
  int tid = blockIdx.x * blockDim.x + threadIdx.x;  // 0..4095 = b*C+c
  int b = tid >> 8;
  int c = tid & (C - 1);
  const float* prevp = out + (size_t)(b * C + c) * T + (size_t)(step - 1) * PATCH;
  float inp[PATCH];
#pragma unroll
  for (int p = 0; p < PATCH; ++p)
    inp[p] = prevp[p] * (g1[c * PATCH + p] * RSQ) + b1[c * PATCH + p];
  float r[PATCH];
#pragma unroll
  for (int pp = 0; pp < PATCH; ++pp) {
    float s = bagg[pp];
#pragma unroll
    for (int j = 0; j < PATCH; ++j) s += inp[j] * Wagg[pp * PATCH + j];
    r[pp] = gelu_exact(s);
  }
  const float* xrow = x + (size_t)(b * C + c) * T + (size_t)step * PATCH;
#pragma unroll
  for (int p = 0; p < PATCH; ++p) {
    int gi = c * T + step * PATCH + p;
    float xt = xrow[p] * (g0[gi] * RSQ) + b0[gi];
    float rv = r[p] + xt;
    res[(b * C + c) * PATCH + p] = rv;
    t[(b * PATCH + p) * C + c] = rv * (g2[c * PATCH + p] * RSQ) + b2[c * PATCH + p];
  }
  if (step + 1 < NPATCH) __builtin_prefetch(xrow + PATCH, 0, 3);  // global_prefetch_b8
}

// per (b,p): top-3 and bottom-3 of t[b,p,:] (values only; ties keep lower index)
__global__ void k_tops(const float* __restrict__ t, float* __restrict__ tops) {
  int bp = threadIdx.x;  // single block of BS*PATCH = 192
  const float* row = t + bp * C;
  float INF = __builtin_inff();
  float M0 = -INF, M1v = -INF, M2v = -INF;
  float m0 = INF, m1 = INF, m2 = INF;
  for (int e = 0; e < C; ++e) {
    float v = row[e];
    if (v > M0)      { M2v = M1v; M1v = M0; M0 = v; }
    else if (v > M1v){ M2v = M1v; M1v = v; }
    else if (v > M2v){ M2v = v; }
    if (v < m0)      { m2 = m1; m1 = m0; m0 = v; }
    else if (v < m1) { m2 = m1; m1 = v; }
    else if (v < m2) { m2 = v; }
  }
  float* o = tops + bp * 6;
  o[0] = M0; o[1] = M1v; o[2] = M2v; o[3] = m0; o[4] = m1; o[5] = m2;
}

// Attention (collapsed to 3-way softmax per head) + msg-MLP via f32 WMMA.
// One wave owns 16 consecutive (b,p,c) rows; A = [16 rows x 4 heads] f32,
// B = M1 [4 x 16 f-cols], C = broadcast base[f]; 16 WMMAs cover FF=256.
__global__ void __launch_bounds__(256)
k_stepC(const float* __restrict__ t, const float* __restrict__ tops,
        const float* __restrict__ res,
        const float* __restrict__ M1g, const float* __restrict__ baseg,
        const float* __restrict__ PRg,
        const float* __restrict__ Wm2, const float* __restrict__ bm2,
        float* __restrict__ out, int step) {
  int gtid = blockIdx.x * blockDim.x + threadIdx.x;
  int wave = gtid >> 5;
  int lane = threadIdx.x & 31;
  int r0 = wave << 4;       // first of 16 rows (row = bp*C + c)
  int bp = r0 >> 8;
  int c0 = r0 & (C - 1);
  int m = lane & 15;        // row within tile (A/M index) and col (N index)
  bool hi = lane >= 16;

  float ph[HEADS], rh[HEADS];
#pragma unroll
  for (int h = 0; h < HEADS; ++h) { ph[h] = PRg[h]; rh[h] = PRg[HEADS + h]; }

  const float* tp = tops + bp * 6;
  float Mx0 = tp[0], Mx1 = tp[1], Mx2 = tp[2];
  float mn0 = tp[3], mn1 = tp[4], mn2 = tp[5];
  const float* trow = t + bp * C;
  float t0 = trow[0], t1 = trow[1], t2 = trow[2];
  float tc = t[r0 + m];

  // per-head neighbor weight w̄_h = softmax3(A*v)·v, branch-free selection
  float wb[HEADS];
#pragma unroll
  for (int h = 0; h < HEADS; ++h) {
    float A = tc * ph[h] + rh[h];
    bool pos = A > 0.0f, neg = A < 0.0f;
    float v0 = pos ? Mx0 : (neg ? mn0 : t0);
    float v1 = pos ? Mx1 : (neg ? mn1 : t1);
    float v2 = pos ? Mx2 : (neg ? mn2 : t2);
    float l0 = A * v0, l1 = A * v1, l2 = A * v2;
    float mx = fmaxf(l0, fmaxf(l1, l2));
    float e0 = expf(l0 - mx), e1 = expf(l1 - mx), e2 = expf(l2 - mx);
    wb[h] = (e0 * v0 + e1 * v1 + e2 * v2) / (e0 + e1 + e2);
  }

  // A-matrix 16x4 f32: lanes 0-15 hold K0/K1, lanes 16-31 hold K2/K3
  v2f a;
  a.x = hi ? wb[2] : wb[0];
  a.y = hi ? wb[3] : wb[1];

  float acc[8];
#pragma unroll
  for (int j = 0; j < 8; ++j) acc[j] = 0.0f;

#pragma unroll
  for (int ft = 0; ft < 16; ++ft) {
    int f = ft * 16 + m;
    // B-matrix 4x16: lanes 0-15 = K0/K1 of col n, lanes 16-31 = K2/K3
    v2f bmat;
    bmat.x = hi ? M1g[2 * FF + f] : M1g[0 * FF + f];
    bmat.y = hi ? M1g[3 * FF + f] : M1g[1 * FF + f];
    float cb = baseg[f];
    v8f cc = {cb, cb, cb, cb, cb, cb, cb, cb};
    v8f d = __builtin_amdgcn_wmma_f32_16x16x4_f32(
        false, a, false, bmat, (short)0, cc, false, false);
    float wm = Wm2[f];
#pragma unroll
    for (int j = 0; j < 8; ++j) acc[j] += gelu_exact(d[j]) * wm;
  }

  // reduce across N within each 16-lane half (xor 1,2,4,8 never crosses bit 4)
#pragma unroll
  for (int j = 0; j < 8; ++j) {
    float s = acc[j];
    s += __shfl_xor(s, 1, 32);
    s += __shfl_xor(s, 2, 32);
    s += __shfl_xor(s, 4, 32);
    s += __shfl_xor(s, 8, 32);
    acc[j] = s;
  }

  // D layout: VGPR j = row j (lanes 0-15) / row j+8 (lanes 16-31)
  if (m == 0) {
    int b = bp / PATCH, p = bp % PATCH;
    int rowBase = hi ? 8 : 0;
    float bias = bm2[0];
#pragma unroll
    for (int j = 0; j < 8; ++j) {
      int c = c0 + rowBase + j;
      float nv = res[(size_t)(b * C + c) * PATCH + p] + ALPHA * (acc[j] + bias);
      out[(size_t)(b * C + c) * T + (size_t)step * PATCH + p] = nv;
    }
  }
}

extern "C" void kernel_launch(void* const* d_in, const int* in_sizes, int n_in,
                              void* d_out, int out_size, void* d_ws, size_t ws_size,
                              hipStream_t stream) {
  (void)in_sizes; (void)n_in; (void)out_size; (void)ws_size;
  const float* x    = (const float*)d_in[0];
  const float* g0   = (const float*)d_in[1];
  const float* b0   = (const float*)d_in[2];
  const float* g1   = (const float*)d_in[3];
  const float* b1   = (const float*)d_in[4];
  const float* g2   = (const float*)d_in[5];
  const float* b2   = (const float*)d_in[6];
  const float* Wagg = (const float*)d_in[7];
  const float* bagg = (const float*)d_in[8];
  const float* We   = (const float*)d_in[9];
  const float* be   = (const float*)d_in[10];
  const float* Wq   = (const float*)d_in[11];
  const float* bq   = (const float*)d_in[12];
  const float* Wk   = (const float*)d_in[13];
  const float* bk   = (const float*)d_in[14];
  const float* Wv   = (const float*)d_in[15];
  const float* bv   = (const float*)d_in[16];
  const float* Wm1  = (const float*)d_in[17];
  const float* bm1  = (const float*)d_in[18];
  const float* Wm2  = (const float*)d_in[19];
  const float* bm2  = (const float*)d_in[20];
  float* out = (float*)d_out;

  float* ws    = (float*)d_ws;
  float* res   = ws;                       // BS*C*PATCH      = 49152
  float* tbuf  = res + BS * C * PATCH;     // BS*PATCH*C      = 49152
  float* tops  = tbuf + BS * PATCH * C;    // BS*PATCH*6      = 1152
  float* M1g   = tops + BS * PATCH * 6;    // HEADS*FF        = 1024
  float* baseg = M1g + HEADS * FF;         // FF              = 256
  float* PRg   = baseg + FF;               // 2*HEADS         = 8

  k_precompute<<<1, FF, 0, stream>>>(We, be, Wq, bq, Wk, bk, Wv, bv, Wm1, bm1,
                                     M1g, baseg, PRg);
  k_init<<<(BS * C * PATCH) / 256, 256, 0, stream>>>(x, g0, b0, out);
  for (int step = 1; step < NPATCH; ++step) {
    k_stepA<<<(BS * C) / 256, 256, 0, stream>>>(x, g0, b0, g1, b1, g2, b2,
                                                Wagg, bagg, out, res, tbuf, step);
    k_tops<<<1, BS * PATCH, 0, stream>>>(tbuf, tops);
    // 3072 waves of 16 rows each: exact fit, EXEC all-1s at every WMMA
    k_stepC<<<(BS * PATCH * C * 2) / 256, 256, 0, stream>>>(
        tbuf, tops, res, M1g, baseg, PRg, Wm2, bm2, out, step);
  }
}